// SparseExpertCountingNetwork_32100585571071
// MI455X (gfx1250) — compile-verified
//
#include <hip/hip_runtime.h>
#include <hip/hip_bf16.h>

typedef float v2f __attribute__((ext_vector_type(2)));
typedef float v4f __attribute__((ext_vector_type(4)));
typedef float v8f __attribute__((ext_vector_type(8)));

#define NROWS 131072
#define HDIM  2048
#define EXP   4
#define TC    32            // tile columns per stage
#define TSTR  36            // padded LDS row stride (floats) -> conflict-free
#define WCOLS 5             // Wt columns: W0..W3, ones (row-sum column)
#define ROWS_PER_WAVE  16
#define WAVES_PER_BLK  8
#define ROWS_PER_BLK   (ROWS_PER_WAVE * WAVES_PER_BLK)   // 128
#define NSTAGES        (HDIM / TC)                       // 64

__global__ __launch_bounds__(256)
void moe_fused_kernel(const float* __restrict__ hist,
                      const float* __restrict__ W,
                      const float* __restrict__ bias,
                      const float* __restrict__ gu,
                      float* __restrict__ out)
{
    // W transposed + ones column: Wt[k][c], c=0..3 -> W[c][k], c=4 -> 1.0. 40 KB.
    __shared__ __attribute__((aligned(16))) float Wt[HDIM * WCOLS];
    // Per-wave staging tile: 16 rows x TSTR floats. 8 * 2304 B = 18 KB.
    __shared__ __attribute__((aligned(16))) float tileMem[WAVES_PER_BLK][ROWS_PER_WAVE * TSTR];

    const int tid  = threadIdx.x;
    const int lane = tid & 31;
    const int wave = tid >> 5;

    // ---- one-time W staging (coalesced per expert row) ----
    for (int k = tid; k < HDIM; k += 256) {
        Wt[k * WCOLS + 0] = W[0 * HDIM + k];
        Wt[k * WCOLS + 1] = W[1 * HDIM + k];
        Wt[k * WCOLS + 2] = W[2 * HDIM + k];
        Wt[k * WCOLS + 3] = W[3 * HDIM + k];
        Wt[k * WCOLS + 4] = 1.0f;            // ones -> C col 4 = row sum
    }
    __syncthreads();

    float* tile = &tileMem[wave][0];
    const int rowBase = blockIdx.x * ROWS_PER_BLK + wave * ROWS_PER_WAVE;

    // WMMA fragment lane constants (16x16x4 f32 layout):
    // lane -> A row / B-C column = lane&15 ; high half handles K+2,K+3.
    const int nn   = lane & 15;
    const int koff = (lane >> 4) * 2;
    // C columns are independent: cols >=5 are never read, so lanes nn>=5 may
    // load ANY finite value -> clamp to the ones column, no selects needed.
    const int cidx = (nn < 4) ? nn : 4;

    // staging lane mapping: 4 rows of 32 cols per load instruction
    const int sgrp = lane >> 3;   // row group 0..3
    const int sl   = lane & 7;    // col group 0..7 (4 floats each)

    v8f c = {0.f, 0.f, 0.f, 0.f, 0.f, 0.f, 0.f, 0.f};
    float mx[4], uq[4], pt[4], carry[4];
#pragma unroll
    for (int i = 0; i < 4; ++i) {
        mx[i] = -3.402823466e38f;
        uq[i] = 0.f;
        pt[i] = 0.f;
        // seed pattern carry with element 0 of this lane's row -> first compare = 0
        carry[i] = hist[(size_t)(rowBase + 4 * i + sgrp) * HDIM];
    }

    for (int s = 0; s < NSTAGES; ++s) {
        const int colBase = s * TC;

        // ---- stage 16x32 tile: coalesced loads + streaming reductions ----
#pragma unroll
        for (int i = 0; i < 4; ++i) {
            const int row = 4 * i + sgrp;
            const float* gp = hist + (size_t)(rowBase + row) * HDIM + colBase + sl * 4;
            v4f x = *(const v4f*)gp;
            if (s + 1 < NSTAGES) __builtin_prefetch(gp + TC, 0, 0);   // global_prefetch_b8

            float pw   = __shfl_up(x.w, 1, 32);
            float prev = (sl == 0) ? carry[i] : pw;                   // row boundary carry
            carry[i]   = __shfl(x.w, lane | 7, 32);                   // last elem of my row seg

            // 3-input max, no NaN-canonicalize traffic (data is NaN-free)
            asm("v_max3_num_f32 %0, %0, %1, %2" : "+v"(mx[i]) : "v"(x.x), "v"(x.y));
            asm("v_max3_num_f32 %0, %0, %1, %2" : "+v"(mx[i]) : "v"(x.z), "v"(x.w));

            uq[i] += (float)(x.x != 0.f) + (float)(x.y != 0.f) +
                     (float)(x.z != 0.f) + (float)(x.w != 0.f);
            pt[i] += (float)(x.x != prev) + (float)(x.y != x.x) +
                     (float)(x.z != x.y) + (float)(x.w != x.z);

            *(v4f*)(tile + row * TSTR + sl * 4) = x;
        }

        // ---- 8 accumulating WMMAs (K = 4 each) over this tile ----
#pragma unroll
        for (int j = 0; j < 8; ++j) {
            const int kG = colBase + 4 * j;   // global K
            const int kL = 4 * j;             // tile-local K
            v2f a = *(const v2f*)(tile + nn * TSTR + kL + koff);
            v2f b;
            b.x = Wt[(kG + koff) * WCOLS + cidx];
            b.y = Wt[(kG + koff + 1) * WCOLS + cidx];
            c = __builtin_amdgcn_wmma_f32_16x16x4_f32(
                    false, a, false, b, (short)0, c, false, false);
        }
    }

    // ---- spill C (logits cols 0..3, row-sum col 4) into dead tile LDS ----
    const int half = lane >> 4;
#pragma unroll
    for (int g = 0; g < 8; ++g) {
        tile[(g + 8 * half) * 20 + nn] = c[g];     // stride 20: aligned + conflict-free
    }

    // ---- reduce per-row streaming stats across 8-lane groups ----
#pragma unroll
    for (int i = 0; i < 4; ++i) {
        float m = mx[i], u = uq[i], p = pt[i];
#pragma unroll
        for (int d = 1; d < 8; d <<= 1) {
            float ms = __shfl_xor(m, d, 32);
            asm("v_max3_num_f32 %0, %0, %1, %1" : "+v"(m) : "v"(ms));
            u += __shfl_xor(u, d, 32);
            p += __shfl_xor(p, d, 32);
        }
        if (sl == 0) {
            const int row = 4 * i + sgrp;
            v4f rv; rv.x = m; rv.y = u; rv.z = p; rv.w = 0.f;
            *(v4f*)(tile + 320 + row * 4) = rv;
        }
    }

    // ---- per-row epilogue: gumbel-softmax routing + expert mix ----
    if (lane < 16) {
        const int row = rowBase + lane;
        v4f lg    = *(const v4f*)(tile + lane * 20);        // logits e=0..3
        float tot = tile[lane * 20 + 4];                    // row sum (WMMA col 4)
        v4f rd    = *(const v4f*)(tile + 320 + lane * 4);   // {max, uniq, patt, -}
        v4f u4    = *(const v4f*)(gu + (size_t)row * EXP);
        v4f bb    = *(const v4f*)bias;

        float z[4];
        float zm = -3.402823466e38f;
#pragma unroll
        for (int e = 0; e < 4; ++e) {
            float uu  = fminf(fmaxf(u4[e], 1e-6f), 1.0f - 1e-6f);
            float gmb = -__logf(-__logf(uu));
            z[e] = lg[e] + bb[e] + gmb;
            zm   = fmaxf(zm, z[e]);
        }
        float ex[4], den = 0.f;
#pragma unroll
        for (int e = 0; e < 4; ++e) { ex[e] = __expf(z[e] - zm); den += ex[e]; }
        const float inv = 1.0f / den;

        const float eo0 = tot;                       // HistogramExpert
        const float eo1 = rd.x / (tot + 1e-6f);      // FrequencyExpert
        const float eo2 = rd.y;                      // UniquenessExpert
        const float eo3 = rd.z;                      // PatternCountExpert

        out[row] = inv * (ex[0] * eo0 + ex[1] * eo1 + ex[2] * eo2 + ex[3] * eo3);
    }
}

extern "C" void kernel_launch(void* const* d_in, const int* in_sizes, int n_in,
                              void* d_out, int out_size, void* d_ws, size_t ws_size,
                              hipStream_t stream) {
    (void)in_sizes; (void)n_in; (void)out_size; (void)d_ws; (void)ws_size;
    const float* hist = (const float*)d_in[0];
    const float* W    = (const float*)d_in[1];
    const float* b    = (const float*)d_in[2];
    const float* gu   = (const float*)d_in[3];
    float*       out  = (float*)d_out;

    dim3 grid(NROWS / ROWS_PER_BLK);   // 1024 workgroups
    dim3 block(256);                   // 8 wave32
    moe_fused_kernel<<<grid, block, 0, stream>>>(hist, W, b, gu, out);
}